// ComplexAttention_27839978012964
// MI455X (gfx1250) — compile-verified
//
#include <hip/hip_runtime.h>
#include <stdint.h>

// ComplexAttention for MI455X (gfx1250, wave32, WMMA f32 16x16x4)
// Flash-style streaming softmax; K/V tiles double-buffered in LDS via
// GLOBAL_LOAD_ASYNC_TO_LDS_B128 (ASYNCcnt), compute fully overlapped with copy.
// Double buffer selected by *integer offset* from the LDS base so every access
// stays in addrspace(3) (ds_load), never generic/flat.
// B,N,L,S,H,E = 4,8,512,512,8,64

typedef __attribute__((ext_vector_type(2))) float v2f;
typedef __attribute__((ext_vector_type(8))) float v8f;

#define B_   4
#define N_   8
#define L_   512
#define S_   512
#define H_   8
#define E_   64
#define ROW  512            // H_*E_  (element stride between consecutive l / s)
#define TS   64             // S-tile staged in LDS
#define EP   68             // padded row stride for K/V tiles (272B rows, 16B aligned)
#define TSP  68             // padded row stride for per-wave attn scratch
#define NW   8              // waves per block
#define BM   128            // NW*16 rows of L per block
#define NT   4              // 16-col tiles per TS
#define KSTEP 16            // E_/4 k-steps for the score GEMM
#define NSTEP (S_ / TS)     // 8 S-tiles
#define MTXSZ (TS * EP)     // floats per staged matrix
#define BUFSZ (4 * MTXSZ)   // floats per (Kr,Ki,Vr,Vi) buffer
#define IMOFF ((size_t)B_ * N_ * L_ * H_ * E_)

// One lane copies 16 bytes global -> LDS on the async path (no data VGPRs).
__device__ __forceinline__ void async_b128(uint32_t lds_addr, const float* gptr) {
    asm volatile("global_load_async_to_lds_b128 %0, %1, off"
                 :: "v"(lds_addr), "v"(gptr)
                 : "memory");
}
__device__ __forceinline__ void wait_async0() {
    asm volatile("s_wait_asynccnt 0" ::: "memory");
}

// Issue the full (Kr,Ki,Vr,Vi) TS x E tile copy for S-tile `st` into the LDS
// buffer starting at byte address `ldsBase`. 4096 b128 chunks over 256 threads.
__device__ __forceinline__ void issue_tile(const float* const* gSrc, uint32_t ldsBase,
                                           int st, int tid) {
    #pragma unroll
    for (int mtx = 0; mtx < 4; ++mtx) {
        #pragma unroll
        for (int r = 0; r < 4; ++r) {
            const int f4 = tid + 256 * r;     // 0..1023 float4 slots per matrix
            const int s  = f4 >> 4;
            const int c4 = f4 & 15;
            const float*   g   = gSrc[mtx] + (size_t)(st * TS + s) * ROW + c4 * 4;
            const uint32_t lds = ldsBase +
                (uint32_t)((mtx * MTXSZ + s * EP + c4 * 4) * sizeof(float));
            async_b128(lds, g);
        }
    }
}

__global__ __launch_bounds__(256) void ca_flash_wmma_f32(
    const float* __restrict__ q_re, const float* __restrict__ q_im,
    const float* __restrict__ k_re, const float* __restrict__ k_im,
    const float* __restrict__ v_re, const float* __restrict__ v_im,
    float* __restrict__ out)
{
    extern __shared__ float smem[];
    // layout: [ buf0 : BUFSZ ][ buf1 : BUFSZ ][ sP : NW*16*TSP ]

    const int tid  = threadIdx.x;
    const int lane = tid & 31;
    const int w    = tid >> 5;              // wave id 0..7
    const int mlo  = lane & 15;
    const int h16  = lane >> 4;             // 0 = lanes 0-15, 1 = lanes 16-31

    const int g     = blockIdx.x >> 2;      // (b*N+n)*H + h
    const int ltile = blockIdx.x & 3;
    const int h     = g & (H_ - 1);
    const int bn    = g >> 3;               // b*N + n

    const size_t qbase  = (size_t)bn * L_ * ROW + (size_t)h * E_;
    const size_t kvbase = (size_t)bn * S_ * ROW + (size_t)h * E_;
    const int    l0     = ltile * BM + w * 16;

    const float* gSrc[4] = { k_re + kvbase, k_im + kvbase, v_re + kvbase, v_im + kvbase };
    const uint32_t ldsBase = (uint32_t)(uintptr_t)smem;      // LDS byte address of buf0
    const int      pOff    = 2 * BUFSZ + w * (16 * TSP);     // this wave's attn scratch

    // Kick off tile 0 copy before touching Q, to hide HBM latency behind Q loads.
    issue_tile(gSrc, ldsBase, 0, tid);

    // ---- Load Q fragments for this wave's 16 rows (A-matrix layout, f32 16x4) ----
    v2f aQr[KSTEP], aQmi[KSTEP];
    {
        const size_t qrow = qbase + (size_t)(l0 + mlo) * ROW;
        #pragma unroll
        for (int k = 0; k < KSTEP; ++k) {
            const int e = k * 4 + 2 * h16;
            aQr[k]      = *(const v2f*)(q_re + qrow + e);
            v2f qi      = *(const v2f*)(q_im + qrow + e);
            aQmi[k]     = -qi;   // fold the re/im subtraction into the WMMA chain
        }
    }

    // ---- Flash state ----
    v8f oRe[NT], oIm[NT];
    #pragma unroll
    for (int f = 0; f < NT; ++f) { oRe[f] = {}; oIm[f] = {}; }
    float mrow[8], lsum[8];
    #pragma unroll
    for (int j = 0; j < 8; ++j) { mrow[j] = -3.0e38f; lsum[j] = 0.0f; }

    wait_async0();
    __syncthreads();

    for (int st = 0; st < NSTEP; ++st) {
        const int cur = (st & 1) * BUFSZ;    // float offset of current buffer
        // matrix base offsets inside smem (all addrspace(3)-rooted GEPs)
        const int oKr = cur;
        const int oKi = cur + MTXSZ;
        const int oVr = cur + 2 * MTXSZ;
        const int oVi = cur + 3 * MTXSZ;

        // Start next tile's copy into the other buffer (safe: its readers
        // finished before the barrier that ended iteration st-1).
        if (st + 1 < NSTEP)
            issue_tile(gSrc,
                       ldsBase + (uint32_t)((((st + 1) & 1) * BUFSZ) * sizeof(float)),
                       st + 1, tid);

        // ---- Scores: s_re = Qr*Kr + (-Qi)*Ki ; s_im = Qr*Ki + (-Qi)*Kr ; |s| ----
        v8f pf[NT];
        #pragma unroll
        for (int nt = 0; nt < NT; ++nt) {
            const int n0 = nt * 16;
            v8f aR = {}, aI = {};
            const int bKr = oKr + (n0 + mlo) * EP + 2 * h16;
            const int bKi = oKi + (n0 + mlo) * EP + 2 * h16;
            #pragma unroll
            for (int k = 0; k < KSTEP; ++k) {
                const v2f bR = *(const v2f*)&smem[bKr + 4 * k];
                const v2f bI = *(const v2f*)&smem[bKi + 4 * k];
                aR = __builtin_amdgcn_wmma_f32_16x16x4_f32(false, aQr[k],  false, bR, (short)0, aR, false, false);
                aR = __builtin_amdgcn_wmma_f32_16x16x4_f32(false, aQmi[k], false, bI, (short)0, aR, false, false);
                aI = __builtin_amdgcn_wmma_f32_16x16x4_f32(false, aQr[k],  false, bI, (short)0, aI, false, false);
                aI = __builtin_amdgcn_wmma_f32_16x16x4_f32(false, aQmi[k], false, bR, (short)0, aI, false, false);
            }
            #pragma unroll
            for (int j = 0; j < 8; ++j)
                pf[nt][j] = sqrtf(aR[j] * aR[j] + aI[j] * aI[j]);
        }

        // ---- Streaming softmax update (rows j / j+8 live in 16-lane halves) ----
        float alpha[8];
        #pragma unroll
        for (int j = 0; j < 8; ++j) {
            float t = fmaxf(fmaxf(pf[0][j], pf[1][j]), fmaxf(pf[2][j], pf[3][j]));
            #pragma unroll
            for (int msk = 1; msk < 16; msk <<= 1)
                t = fmaxf(t, __shfl_xor(t, msk, 32));
            const float mn = fmaxf(mrow[j], t);
            alpha[j] = __expf(mrow[j] - mn);
            mrow[j]  = mn;
        }
        float rs[8];
        #pragma unroll
        for (int j = 0; j < 8; ++j) rs[j] = 0.0f;
        #pragma unroll
        for (int nt = 0; nt < NT; ++nt) {
            #pragma unroll
            for (int j = 0; j < 8; ++j) {
                const float p = __expf(pf[nt][j] - mrow[j]);
                pf[nt][j] = p;
                rs[j] += p;
            }
        }
        #pragma unroll
        for (int j = 0; j < 8; ++j) {
            float t = rs[j];
            #pragma unroll
            for (int msk = 1; msk < 16; msk <<= 1)
                t += __shfl_xor(t, msk, 32);
            lsum[j] = lsum[j] * alpha[j] + t;
        }
        #pragma unroll
        for (int f = 0; f < NT; ++f) {
            #pragma unroll
            for (int j = 0; j < 8; ++j) { oRe[f][j] *= alpha[j]; oIm[f][j] *= alpha[j]; }
        }

        // ---- C/D-layout attn tile -> LDS -> A-matrix layout for attn @ V ----
        #pragma unroll
        for (int nt = 0; nt < NT; ++nt) {
            #pragma unroll
            for (int j = 0; j < 8; ++j)
                smem[pOff + (j + 8 * h16) * TSP + nt * 16 + mlo] = pf[nt][j];
        }
        // per-wave scratch: same-wave DS ops are in-order; compiler inserts dscnt waits

        #pragma unroll
        for (int f = 0; f < NT; ++f) {
            const int ec = f * 16 + mlo;
            #pragma unroll
            for (int k = 0; k < TS / 4; ++k) {
                const int s0 = k * 4 + 2 * h16;
                const v2f aP  = *(const v2f*)&smem[pOff + mlo * TSP + s0];
                const v2f bVr = { smem[oVr + s0 * EP + ec], smem[oVr + (s0 + 1) * EP + ec] };
                const v2f bVi = { smem[oVi + s0 * EP + ec], smem[oVi + (s0 + 1) * EP + ec] };
                oRe[f] = __builtin_amdgcn_wmma_f32_16x16x4_f32(false, aP, false, bVr, (short)0, oRe[f], false, false);
                oIm[f] = __builtin_amdgcn_wmma_f32_16x16x4_f32(false, aP, false, bVi, (short)0, oIm[f], false, false);
            }
        }

        // Next buffer ready + everyone done reading current buffer.
        if (st + 1 < NSTEP) {
            wait_async0();
            __syncthreads();
        }
    }

    // ---- Normalize and store [2,b,n,l,h,e] ----
    const size_t obase = (size_t)bn * L_ * ROW + (size_t)h * E_;
    float inv[8];
    #pragma unroll
    for (int j = 0; j < 8; ++j) inv[j] = 1.0f / lsum[j];
    #pragma unroll
    for (int f = 0; f < NT; ++f) {
        #pragma unroll
        for (int j = 0; j < 8; ++j) {
            const int row = j + 8 * h16;
            const size_t o = obase + (size_t)(l0 + row) * ROW + f * 16 + mlo;
            out[o]         = oRe[f][j] * inv[j];
            out[o + IMOFF] = oIm[f][j] * inv[j];
        }
    }
}

extern "C" void kernel_launch(void* const* d_in, const int* in_sizes, int n_in,
                              void* d_out, int out_size, void* d_ws, size_t ws_size,
                              hipStream_t stream) {
    (void)in_sizes; (void)n_in; (void)out_size; (void)d_ws; (void)ws_size;
    const float* q_re = (const float*)d_in[0];
    const float* q_im = (const float*)d_in[1];
    const float* k_re = (const float*)d_in[2];
    const float* k_im = (const float*)d_in[3];
    const float* v_re = (const float*)d_in[4];
    const float* v_im = (const float*)d_in[5];
    float* out = (float*)d_out;

    const dim3 grid(B_ * N_ * H_ * (L_ / BM));   // 1024 blocks
    const dim3 block(256);                       // 8 waves (wave32)
    const size_t smem = (size_t)(2 * BUFSZ + NW * 16 * TSP) * sizeof(float); // ~170 KB

    ca_flash_wmma_f32<<<grid, block, smem, stream>>>(q_re, q_im, k_re, k_im,
                                                     v_re, v_im, out);
}